// Qwen3MoeSparseMoeBlock_17970143166636
// MI455X (gfx1250) — compile-verified
//
#include <hip/hip_runtime.h>
#include <hip/hip_bf16.h>

#define H_DIM 2048
#define I_DIM 768
#define E_NUM 32
#define TOPK 4
#define T_NUM 2048

#define TILE_M 128
#define TILE_N 64
#define TILE_K 32
#define A_PAD 4  // row stride 36 floats: 16B-aligned rows, conflict-free columns

typedef __attribute__((ext_vector_type(2))) float v2f;
typedef __attribute__((ext_vector_type(8))) float v8f;

// ---------------------------------------------------------------------------
// gfx1250 async global->LDS copy (ASYNCcnt path, no VGPR round trip).
// VDST = 32-bit LDS byte address (generic LDS pointer truncates to it),
// VADDR = 64-bit global address, SADDR = off.
// ---------------------------------------------------------------------------
__device__ __forceinline__ void async_copy_b128(const void* gptr,
                                                const void* lptr) {
  unsigned lds_addr = (unsigned)(size_t)lptr;
  asm volatile("global_load_async_to_lds_b128 %0, %1, off" ::"v"(lds_addr),
               "v"(gptr)
               : "memory");
}
__device__ __forceinline__ void wait_async_all() {
  asm volatile("s_wait_asynccnt 0x0" ::: "memory");
}

// ---------------------------------------------------------------------------
// Kernel 1: router. One wave32 per token; lane == expert (E == 32 == warpSize).
// logits = x @ gate_w^T ; softmax over experts ; top-4 ; renormalize.
// ---------------------------------------------------------------------------
__global__ __launch_bounds__(256) void router_kernel(
    const float* __restrict__ x, const float* __restrict__ gate_w,
    int* __restrict__ topk_idx, float* __restrict__ topk_w) {
  const int wave = threadIdx.x >> 5;
  const int lane = threadIdx.x & 31;
  const int t = blockIdx.x * (blockDim.x >> 5) + wave;
  if (t >= T_NUM) return;

  const float* xr = x + (size_t)t * H_DIM;
  const float* gr = gate_w + (size_t)lane * H_DIM;  // lane = expert id
  float acc = 0.f;
  for (int h = 0; h < H_DIM; h += 4) {
    float4 xv = *(const float4*)(xr + h);
    float4 gv = *(const float4*)(gr + h);
    acc += xv.x * gv.x + xv.y * gv.y + xv.z * gv.z + xv.w * gv.w;
  }
  // softmax across the 32 lanes
  float m = acc;
  for (int off = 16; off >= 1; off >>= 1) m = fmaxf(m, __shfl_xor(m, off, 32));
  float p = __expf(acc - m);
  float s = p;
  for (int off = 16; off >= 1; off >>= 1) s += __shfl_xor(s, off, 32);
  float prob = p / s;

  // top-4 via repeated argmax reduction (tie -> lowest index, matches lax.top_k)
  float wk[TOPK];
  int ik[TOPK];
  float wsum = 0.f;
  float cur = prob;
  for (int k = 0; k < TOPK; ++k) {
    float v = cur;
    int idx = lane;
    for (int off = 16; off >= 1; off >>= 1) {
      float v2 = __shfl_xor(v, off, 32);
      int i2 = __shfl_xor(idx, off, 32);
      if (v2 > v || (v2 == v && i2 < idx)) { v = v2; idx = i2; }
    }
    wk[k] = v;
    ik[k] = idx;
    wsum += v;
    if (lane == idx) cur = -1.f;  // exclude from next round
  }
  if (lane < TOPK) {
    topk_w[t * TOPK + lane] = wk[lane] / wsum;
    topk_idx[t * TOPK + lane] = ik[lane];
  }
}

// ---------------------------------------------------------------------------
// Kernel 2: deterministic per-expert token-list build (one wave per expert).
// Order-preserving compaction via ballot+popc -> bitwise-stable lists.
// Entry encodes (token << 2) | slot.
// ---------------------------------------------------------------------------
__global__ void build_lists_kernel(const int* __restrict__ topk_idx,
                                   int* __restrict__ pair_tok,
                                   int* __restrict__ counts) {
  const int e = blockIdx.x;
  const int lane = threadIdx.x;  // blockDim == 32
  int cnt = 0;
  for (int base = 0; base < T_NUM; base += 32) {
    int t = base + lane;
    int slot = -1;
#pragma unroll
    for (int k = 0; k < TOPK; ++k)
      if (topk_idx[t * TOPK + k] == e) slot = k;
    unsigned mask = (unsigned)__ballot(slot >= 0);
    if (slot >= 0) {
      int pos = cnt + __popc(mask & ((1u << lane) - 1u));
      pair_tok[e * T_NUM + pos] = (t << 2) | slot;
    }
    cnt += __popc(mask);
  }
  if (lane == 0) counts[e] = cnt;
}

// ---------------------------------------------------------------------------
// Kernel 3: exclusive prefix sum over the 32 expert counts (single wave).
// ---------------------------------------------------------------------------
__global__ void offsets_kernel(const int* __restrict__ counts,
                               int* __restrict__ offs) {
  const int lane = threadIdx.x;
  int c = counts[lane];
  int xv = c;
  for (int off = 1; off < 32; off <<= 1) {
    int y = __shfl_up(xv, off, 32);
    if (lane >= off) xv += y;
  }
  offs[lane] = xv - c;  // exclusive
}

// ---------------------------------------------------------------------------
// Kernel 4: gathered gate+up GEMM per expert with fused SiLU*up*route-weight.
// Block tile: 128 (pair rows) x 64 (of I), 8 waves, K over H in 32-chunks.
// Double-buffered LDS fed by async global->LDS B128 copies; each wave owns a
// 16-row band and 4 N-subtiles of v_wmma_f32_16x16x4_f32 accumulators.
// ---------------------------------------------------------------------------
__global__ __launch_bounds__(256) void moe_gateup_kernel(
    const float* __restrict__ x, const float* __restrict__ Wg,
    const float* __restrict__ Wu, const int* __restrict__ pair_tok,
    const float* __restrict__ topk_w, const int* __restrict__ counts,
    const int* __restrict__ offs, float* __restrict__ act) {
  const int e = blockIdx.x;
  const int cnt = counts[e];
  const int m0 = blockIdx.y * TILE_M;
  if (m0 >= cnt) return;  // uniform exit: EXEC stays all-ones for WMMA
  const int n0 = blockIdx.z * TILE_N;

  __shared__ float As[2][TILE_M][TILE_K + A_PAD];
  __shared__ float Bg[2][TILE_K][TILE_N];
  __shared__ float Bu[2][TILE_K][TILE_N];
  __shared__ int toks[TILE_M];
  __shared__ float wrow[TILE_M];

  const int tid = threadIdx.x;
  const int lane = tid & 31;
  const int wave = tid >> 5;

  if (tid < TILE_M) {
    int mm = m0 + tid;
    int pt = (mm < cnt) ? pair_tok[e * T_NUM + mm] : 0;
    int t = pt >> 2, slot = pt & 3;
    toks[tid] = t;
    wrow[tid] = (mm < cnt) ? topk_w[t * TOPK + slot] : 0.f;
  }
  __syncthreads();

  // Per-thread fixed copy slots (addresses precomputed; stage adds h0 terms).
  const float* Wge = Wg + (size_t)e * H_DIM * I_DIM;
  const float* Wue = Wu + (size_t)e * H_DIM * I_DIM;
  const float* gA[4];
  int rA[4], cA[4];
#pragma unroll
  for (int p = 0; p < 4; ++p) {
    int idx = p * 256 + tid;  // b128 slot: 8 per 32-float row
    rA[p] = idx >> 3;
    cA[p] = (idx & 7) * 4;
    gA[p] = x + (size_t)toks[rA[p]] * H_DIM + cA[p];
  }
  const float* gBg[2];
  const float* gBu[2];
  int kB[2], cB[2];
#pragma unroll
  for (int p = 0; p < 2; ++p) {
    int idx = p * 256 + tid;  // b128 slot: 16 per 64-float row
    kB[p] = idx >> 4;
    cB[p] = (idx & 15) * 4;
    gBg[p] = Wge + (size_t)kB[p] * I_DIM + n0 + cB[p];
    gBu[p] = Wue + (size_t)kB[p] * I_DIM + n0 + cB[p];
  }

  v8f accG[4], accU[4];
#pragma unroll
  for (int s2 = 0; s2 < 4; ++s2)
#pragma unroll
    for (int r = 0; r < 8; ++r) { accG[s2][r] = 0.f; accU[s2][r] = 0.f; }

  // Prologue: async-stage chunk 0 into buffer 0.
#pragma unroll
  for (int p = 0; p < 4; ++p)
    async_copy_b128(gA[p], &As[0][rA[p]][cA[p]]);
#pragma unroll
  for (int p = 0; p < 2; ++p) {
    async_copy_b128(gBg[p], &Bg[0][kB[p]][cB[p]]);
    async_copy_b128(gBu[p], &Bu[0][kB[p]][cB[p]]);
  }

  int cur = 0;
  for (int h0 = 0; h0 < H_DIM; h0 += TILE_K) {
    wait_async_all();   // my stage-h0 copies have landed in LDS
    __syncthreads();    // everyone's landed; prev reads of other buffer done

    if (h0 + TILE_K < H_DIM) {
      const int nb = cur ^ 1;
      const int hn = h0 + TILE_K;
#pragma unroll
      for (int p = 0; p < 4; ++p)
        async_copy_b128(gA[p] + hn, &As[nb][rA[p]][cA[p]]);
#pragma unroll
      for (int p = 0; p < 2; ++p) {
        async_copy_b128(gBg[p] + (size_t)hn * I_DIM, &Bg[nb][kB[p]][cB[p]]);
        async_copy_b128(gBu[p] + (size_t)hn * I_DIM, &Bu[nb][kB[p]][cB[p]]);
      }
    }

    const int rowA = wave * 16 + (lane & 15);
    const int hi = lane >> 4;
#pragma unroll
    for (int kk = 0; kk < TILE_K; kk += 4) {
      v2f a;
      a.x = As[cur][rowA][kk + 2 * hi];
      a.y = As[cur][rowA][kk + 2 * hi + 1];
#pragma unroll
      for (int s2 = 0; s2 < 4; ++s2) {
        int n = s2 * 16 + (lane & 15);
        v2f bg, bu;
        bg.x = Bg[cur][kk + 2 * hi][n]; bg.y = Bg[cur][kk + 2 * hi + 1][n];
        bu.x = Bu[cur][kk + 2 * hi][n]; bu.y = Bu[cur][kk + 2 * hi + 1][n];
        accG[s2] = __builtin_amdgcn_wmma_f32_16x16x4_f32(
            false, a, false, bg, (short)0, accG[s2], false, false);
        accU[s2] = __builtin_amdgcn_wmma_f32_16x16x4_f32(
            false, a, false, bu, (short)0, accU[s2], false, false);
      }
    }
    cur ^= 1;
  }

  // Epilogue: a = silu(g) * u * route_weight -> packed act buffer
  const int off_e = offs[e];
  const int hi2 = lane >> 4;
#pragma unroll
  for (int s2 = 0; s2 < 4; ++s2) {
#pragma unroll
    for (int r = 0; r < 8; ++r) {
      int rl = wave * 16 + r + 8 * hi2;  // C layout: M = r + 8*(lane>=16)
      int mm = m0 + rl;
      if (mm < cnt) {
        int col = n0 + s2 * 16 + (lane & 15);
        float g = accG[s2][r], u = accU[s2][r];
        float a = g * (1.f / (1.f + __expf(-g))) * u * wrow[rl];
        act[(size_t)(off_e + mm) * I_DIM + col] = a;
      }
    }
  }
}

// ---------------------------------------------------------------------------
// Kernel 5: down projection (act @ Wd[e]) with scatter-add into out[token].
// Same double-buffered async-LDS pipeline; padded rows clamp to a valid act
// row (their C rows are never stored).
// ---------------------------------------------------------------------------
__global__ __launch_bounds__(256) void moe_down_kernel(
    const float* __restrict__ act, const float* __restrict__ Wd,
    const int* __restrict__ pair_tok, const int* __restrict__ counts,
    const int* __restrict__ offs, float* __restrict__ out) {
  const int e = blockIdx.x;
  const int cnt = counts[e];
  const int m0 = blockIdx.y * TILE_M;
  if (m0 >= cnt) return;  // uniform exit
  const int n0 = blockIdx.z * TILE_N;  // over H

  __shared__ float As[2][TILE_M][TILE_K + A_PAD];
  __shared__ float Bs[2][TILE_K][TILE_N];
  __shared__ int toks[TILE_M];

  const int tid = threadIdx.x;
  const int lane = tid & 31;
  const int wave = tid >> 5;
  const int off_e = offs[e];

  if (tid < TILE_M) {
    int mm = m0 + tid;
    toks[tid] = (mm < cnt) ? (pair_tok[e * T_NUM + mm] >> 2) : 0;
  }
  __syncthreads();

  const float* Wde = Wd + (size_t)e * I_DIM * H_DIM;
  const float* gA[4];
  int rA[4], cA[4];
#pragma unroll
  for (int p = 0; p < 4; ++p) {
    int idx = p * 256 + tid;
    rA[p] = idx >> 3;
    cA[p] = (idx & 7) * 4;
    int mm = m0 + rA[p];
    int row = off_e + ((mm < cnt) ? mm : (cnt - 1));  // clamp: stay in-bounds
    gA[p] = act + (size_t)row * I_DIM + cA[p];
  }
  const float* gB[2];
  int kB[2], cB[2];
#pragma unroll
  for (int p = 0; p < 2; ++p) {
    int idx = p * 256 + tid;
    kB[p] = idx >> 4;
    cB[p] = (idx & 15) * 4;
    gB[p] = Wde + (size_t)kB[p] * H_DIM + n0 + cB[p];
  }

  v8f acc[4];
#pragma unroll
  for (int s2 = 0; s2 < 4; ++s2)
#pragma unroll
    for (int r = 0; r < 8; ++r) acc[s2][r] = 0.f;

  // Prologue: stage chunk 0 into buffer 0.
#pragma unroll
  for (int p = 0; p < 4; ++p)
    async_copy_b128(gA[p], &As[0][rA[p]][cA[p]]);
#pragma unroll
  for (int p = 0; p < 2; ++p)
    async_copy_b128(gB[p], &Bs[0][kB[p]][cB[p]]);

  int cur = 0;
  for (int i0 = 0; i0 < I_DIM; i0 += TILE_K) {  // 24 K-chunks
    wait_async_all();
    __syncthreads();

    if (i0 + TILE_K < I_DIM) {
      const int nb = cur ^ 1;
      const int in = i0 + TILE_K;
#pragma unroll
      for (int p = 0; p < 4; ++p)
        async_copy_b128(gA[p] + in, &As[nb][rA[p]][cA[p]]);
#pragma unroll
      for (int p = 0; p < 2; ++p)
        async_copy_b128(gB[p] + (size_t)in * H_DIM, &Bs[nb][kB[p]][cB[p]]);
    }

    const int rowA = wave * 16 + (lane & 15);
    const int hi = lane >> 4;
#pragma unroll
    for (int kk = 0; kk < TILE_K; kk += 4) {
      v2f a;
      a.x = As[cur][rowA][kk + 2 * hi];
      a.y = As[cur][rowA][kk + 2 * hi + 1];
#pragma unroll
      for (int s2 = 0; s2 < 4; ++s2) {
        int n = s2 * 16 + (lane & 15);
        v2f b;
        b.x = Bs[cur][kk + 2 * hi][n]; b.y = Bs[cur][kk + 2 * hi + 1][n];
        acc[s2] = __builtin_amdgcn_wmma_f32_16x16x4_f32(
            false, a, false, b, (short)0, acc[s2], false, false);
      }
    }
    cur ^= 1;
  }

  // Scatter-add into output tokens (hardware fp32 global atomic add)
  const int hi2 = lane >> 4;
#pragma unroll
  for (int s2 = 0; s2 < 4; ++s2) {
#pragma unroll
    for (int r = 0; r < 8; ++r) {
      int rl = wave * 16 + r + 8 * hi2;
      int mm = m0 + rl;
      if (mm < cnt) {
        int col = n0 + s2 * 16 + (lane & 15);
        float* dst = out + (size_t)toks[rl] * H_DIM + col;
        __hip_atomic_fetch_add(dst, acc[s2][r], __ATOMIC_RELAXED,
                               __HIP_MEMORY_SCOPE_AGENT);
      }
    }
  }
}

// ---------------------------------------------------------------------------
// Launch
// ---------------------------------------------------------------------------
extern "C" void kernel_launch(void* const* d_in, const int* in_sizes, int n_in,
                              void* d_out, int out_size, void* d_ws,
                              size_t ws_size, hipStream_t stream) {
  const float* x = (const float*)d_in[0];       // [1, 2048, 2048]
  const float* gate_w = (const float*)d_in[1];  // [E, H]
  const float* Wg = (const float*)d_in[2];      // [E, H, I]
  const float* Wu = (const float*)d_in[3];      // [E, H, I]
  const float* Wd = (const float*)d_in[4];      // [E, I, H]
  // d_in[5] = top_k (compile-time TOPK == 4)

  // Workspace carve (all 4-byte types; act offset is 16B aligned).
  int* topk_idx = (int*)d_ws;                         // [T*K]
  float* topk_w = (float*)(topk_idx + T_NUM * TOPK);  // [T*K]
  int* counts = (int*)(topk_w + T_NUM * TOPK);        // [E]
  int* offs = counts + E_NUM;                         // [E]
  int* pair_tok = offs + E_NUM;                       // [E*T] capacity
  float* act = (float*)(pair_tok + E_NUM * T_NUM);    // [T*K * I] ~ 25 MB

  hipMemsetAsync(d_out, 0, (size_t)out_size * sizeof(float), stream);

  router_kernel<<<T_NUM / 8, 256, 0, stream>>>(x, gate_w, topk_idx, topk_w);
  build_lists_kernel<<<E_NUM, 32, 0, stream>>>(topk_idx, pair_tok, counts);
  offsets_kernel<<<1, 32, 0, stream>>>(counts, offs);

  dim3 g1(E_NUM, T_NUM / TILE_M, I_DIM / TILE_N);  // (32, 16, 12)
  moe_gateup_kernel<<<g1, 256, 0, stream>>>(x, Wg, Wu, pair_tok, topk_w,
                                            counts, offs, act);

  dim3 g2(E_NUM, T_NUM / TILE_M, H_DIM / TILE_N);  // (32, 16, 32)
  moe_down_kernel<<<g2, 256, 0, stream>>>(act, Wd, pair_tok, counts, offs,
                                          (float*)d_out);
}